// GATConv_35244501631115
// MI455X (gfx1250) — compile-verified
//
#include <hip/hip_runtime.h>
#include <hip/hip_bf16.h>

typedef float v2f __attribute__((ext_vector_type(2)));
typedef float v4f __attribute__((ext_vector_type(4)));
typedef float v8f __attribute__((ext_vector_type(8)));

#define NODES_PER_TILE 16
#define HD 64            // H * OUT_DIM
#define NH 4             // heads
#define OD 16            // out dim per head
#define INDIM 128
#define EDIM 32
#define NEG_SLOPE 0.2f

// ---------- monotonic float <-> uint key for atomic segment-max ----------
__device__ __forceinline__ unsigned fkey(float f) {
    unsigned u = __float_as_uint(f);
    return (u & 0x80000000u) ? ~u : (u | 0x80000000u);
}
__device__ __forceinline__ float unfkey(unsigned k) {
    unsigned u = (k & 0x80000000u) ? (k & 0x7FFFFFFFu) : ~k;
    return __uint_as_float(u);
}
#define KEY_NEG_INF 0x007FFFFFu   // fkey(-inf)

// ---------- kernel 1: fold edge weight: w_e[h][k] = sum_d W_edge[h*16+d][k]*attn_e[h][d]
__global__ void k_fold_we(const float* __restrict__ W_edge,
                          const float* __restrict__ attn_e,
                          float* __restrict__ we) {
    int t = threadIdx.x;           // 128 threads: h = t/32, k = t%32
    if (t >= NH * EDIM) return;
    int h = t >> 5, k = t & 31;
    float s = 0.f;
    #pragma unroll
    for (int d = 0; d < OD; ++d)
        s += W_edge[(h * OD + d) * EDIM + k] * attn_e[h * OD + d];
    we[h * EDIM + k] = s;
}

// ---------- kernel 2: node projection feat_ = feat @ W_fc^T via V_WMMA_F32_16X16X4_F32
// one wave computes a 16-node x 64-col tile; 4 accumulators (one per head/16-col tile)
__global__ void k_proj_wmma(const float* __restrict__ feat,
                            const float* __restrict__ Wfc,
                            float* __restrict__ featp, int n) {
    int wavesPerBlock = blockDim.x >> 5;
    int wave = blockIdx.x * wavesPerBlock + (threadIdx.x >> 5);
    int lane = threadIdx.x & 31;
    int row0 = wave * NODES_PER_TILE;
    if (row0 >= n) return;                    // wave-uniform exit

    int m  = lane & 15;                       // A row / B-D column within tile
    int kh = lane >> 4;                       // K half: lanes 0-15 -> K{0,1}, 16-31 -> K{2,3}

    int arow = row0 + m; if (arow >= n) arow = n - 1;   // clamp (N%16==0 normally)
    const float* ap = feat + (size_t)arow * INDIM + 2 * kh;

    v8f acc[4];
    #pragma unroll
    for (int t = 0; t < 4; ++t) acc[t] = (v8f){0,0,0,0,0,0,0,0};

    for (int k0 = 0; k0 < INDIM; k0 += 4) {
        v2f a = *(const v2f*)(ap + k0);       // A: feat[row][k0+2kh .. +1]
        #pragma unroll
        for (int t = 0; t < 4; ++t) {
            // B[k][ncol] = Wfc[ncol][k], ncol = t*16 + m
            v2f b = *(const v2f*)(Wfc + (size_t)(t * OD + m) * INDIM + k0 + 2 * kh);
            acc[t] = __builtin_amdgcn_wmma_f32_16x16x4_f32(
                false, a, false, b, (short)0, acc[t], false, false);
        }
    }

    // C/D layout: VGPR r of lane l holds M = r + 8*(l/16), Ncol = l%16
    float* outp = featp + (size_t)(row0 + 8 * kh) * HD + m;
    if (row0 + NODES_PER_TILE <= n) {
        // full tile: unguarded coalesced stores (16 lanes -> 64B per store)
        #pragma unroll
        for (int t = 0; t < 4; ++t) {
            #pragma unroll
            for (int r = 0; r < 8; ++r)
                outp[(size_t)r * HD + t * OD] = acc[t][r];
        }
    } else {
        #pragma unroll
        for (int t = 0; t < 4; ++t) {
            #pragma unroll
            for (int r = 0; r < 8; ++r) {
                int row = row0 + r + 8 * kh;
                if (row < n) featp[(size_t)row * HD + t * OD + m] = acc[t][r];
            }
        }
    }
}

// ---------- kernel 3: eh/et per-node attention scores (float4 loads)
__global__ void k_node_scores(const float* __restrict__ featp,
                              const float* __restrict__ attn_h,
                              const float* __restrict__ attn_t,
                              float* __restrict__ eh, float* __restrict__ et, int n) {
    int nid = blockIdx.x * blockDim.x + threadIdx.x;
    if (nid >= n) return;
    const v4f* fp = (const v4f*)(featp + (size_t)nid * HD);
    const v4f* ah = (const v4f*)attn_h;
    const v4f* at = (const v4f*)attn_t;
    #pragma unroll
    for (int h = 0; h < NH; ++h) {
        float sh = 0.f, st = 0.f;
        #pragma unroll
        for (int q = 0; q < OD / 4; ++q) {
            v4f v = fp[h * 4 + q];
            v4f wh = ah[h * 4 + q];
            v4f wt = at[h * 4 + q];
            sh += v.x * wh.x + v.y * wh.y + v.z * wh.z + v.w * wh.w;
            st += v.x * wt.x + v.y * wt.y + v.z * wt.z + v.w * wt.w;
        }
        eh[nid * NH + h] = sh;
        et[nid * NH + h] = st;
    }
}

// ---------- kernel 4: init out=bias, emax keys, asum
__global__ void k_init(float* __restrict__ out, const float* __restrict__ bias,
                       unsigned* __restrict__ emaxkey, float* __restrict__ asum, int n) {
    int idx = blockIdx.x * blockDim.x + threadIdx.x;
    if (idx < n * HD) out[idx] = bias[idx & (HD - 1)];
    if (idx < n * NH) { emaxkey[idx] = KEY_NEG_INF; asum[idx] = 0.f; }
}

// ---------- kernel 5: per-edge logits (leaky relu) + atomic segment max over dst
__global__ void k_edge_logits(const float* __restrict__ ea,
                              const int* __restrict__ src, const int* __restrict__ dst,
                              const float* __restrict__ eh, const float* __restrict__ et,
                              const float* __restrict__ we,
                              float* __restrict__ logit, unsigned* __restrict__ emaxkey,
                              int e) {
    __shared__ float swe[NH * EDIM];
    if (threadIdx.x < NH * EDIM) swe[threadIdx.x] = we[threadIdx.x];
    __syncthreads();
    int eid = blockIdx.x * blockDim.x + threadIdx.x;
    if (eid >= e) return;
    const v4f* ap = (const v4f*)(ea + (size_t)eid * EDIM);   // 8 x float4 (b128 loads)
    float acc[NH] = {0.f, 0.f, 0.f, 0.f};
    #pragma unroll
    for (int q = 0; q < EDIM / 4; ++q) {
        v4f v = ap[q];
        #pragma unroll
        for (int h = 0; h < NH; ++h) {
            acc[h] += v.x * swe[h * EDIM + 4 * q + 0];
            acc[h] += v.y * swe[h * EDIM + 4 * q + 1];
            acc[h] += v.z * swe[h * EDIM + 4 * q + 2];
            acc[h] += v.w * swe[h * EDIM + 4 * q + 3];
        }
    }
    int s = src[eid], d = dst[eid];
    #pragma unroll
    for (int h = 0; h < NH; ++h) {
        float l = eh[s * NH + h] + acc[h] + et[d * NH + h];
        l = l > 0.f ? l : NEG_SLOPE * l;
        logit[(size_t)eid * NH + h] = l;
        atomicMax(&emaxkey[d * NH + h], fkey(l));
    }
}

// ---------- kernel 6: a = exp(logit - max[dst]); atomic segment sum
__global__ void k_edge_exp(const int* __restrict__ dst,
                           const unsigned* __restrict__ emaxkey,
                           float* __restrict__ logit, float* __restrict__ asum, int e) {
    int eid = blockIdx.x * blockDim.x + threadIdx.x;
    if (eid >= e) return;
    int d = dst[eid];
    #pragma unroll
    for (int h = 0; h < NH; ++h) {
        float mx = unfkey(emaxkey[d * NH + h]);   // finite: this dst has >=1 edge
        float a = __expf(logit[(size_t)eid * NH + h] - mx);
        logit[(size_t)eid * NH + h] = a;
        atomicAdd(&asum[d * NH + h], a);
    }
}

// ---------- kernel 7: asum -> reciprocal (hoist divides out of the hot scatter)
__global__ void k_asum_recip(float* __restrict__ asum, int n4) {
    int i = blockIdx.x * blockDim.x + threadIdx.x;
    if (i >= n4) return;
    float s = asum[i];
    asum[i] = (s > 0.f) ? (1.0f / s) : 1.0f;
}

// ---------- kernel 8: aggregation; 64 threads per edge, coalesced gather + float atomics
__global__ void k_aggregate(const int* __restrict__ src, const int* __restrict__ dst,
                            const float* __restrict__ a, const float* __restrict__ rinv,
                            const float* __restrict__ featp, float* __restrict__ out,
                            int e) {
    int gid = blockIdx.x * blockDim.x + threadIdx.x;
    int eid = gid >> 6;
    if (eid >= e) return;
    int hd = gid & (HD - 1);
    int h  = hd >> 4;
    int s = src[eid], d = dst[eid];
    float w = a[(size_t)eid * NH + h] * rinv[d * NH + h];
    float val = featp[(size_t)s * HD + hd] * w;
    atomicAdd(&out[(size_t)d * HD + hd], val);
}

extern "C" void kernel_launch(void* const* d_in, const int* in_sizes, int n_in,
                              void* d_out, int out_size, void* d_ws, size_t ws_size,
                              hipStream_t stream) {
    const float* feat      = (const float*)d_in[0];   // [N,128]
    const float* edge_attr = (const float*)d_in[1];   // [E,32]
    const int*   src       = (const int*)d_in[2];     // [E]
    const int*   dst       = (const int*)d_in[3];     // [E]
    const float* W_fc      = (const float*)d_in[4];   // [64,128]
    const float* W_edge    = (const float*)d_in[5];   // [64,32]
    const float* attn_h    = (const float*)d_in[6];   // [4,16]
    const float* attn_t    = (const float*)d_in[7];   // [4,16]
    const float* attn_e    = (const float*)d_in[8];   // [4,16]
    const float* bias      = (const float*)d_in[9];   // [4,16]
    float* out = (float*)d_out;                       // [N,64]

    const int N = in_sizes[0] / INDIM;
    const int E = in_sizes[2];

    // workspace layout (floats)
    char* ws = (char*)d_ws;
    float*    featp   = (float*)ws;                            ws += (size_t)N * HD * 4;
    float*    eh      = (float*)ws;                            ws += (size_t)N * NH * 4;
    float*    et      = (float*)ws;                            ws += (size_t)N * NH * 4;
    float*    asum    = (float*)ws;                            ws += (size_t)N * NH * 4;
    unsigned* emaxkey = (unsigned*)ws;                         ws += (size_t)N * NH * 4;
    float*    we      = (float*)ws;                            ws += (size_t)NH * EDIM * 4;
    float*    logit   = (float*)ws;                            ws += (size_t)E * NH * 4;

    const int B = 256;

    // 1. fold edge weights
    k_fold_we<<<1, 128, 0, stream>>>(W_edge, attn_e, we);

    // 2. WMMA node projection: one wave per 16-row tile, 8 waves/block
    int tiles  = (N + NODES_PER_TILE - 1) / NODES_PER_TILE;
    int blocks = (tiles + 7) / 8;
    k_proj_wmma<<<blocks, 256, 0, stream>>>(feat, W_fc, featp, N);

    // 3. per-node scores
    k_node_scores<<<(N + B - 1) / B, B, 0, stream>>>(featp, attn_h, attn_t, eh, et, N);

    // 4. init out=bias, max keys, sums
    k_init<<<((size_t)N * HD + B - 1) / B, B, 0, stream>>>(out, bias, emaxkey, asum, N);

    // 5. edge logits + segment max
    k_edge_logits<<<(E + B - 1) / B, B, 0, stream>>>(edge_attr, src, dst, eh, et, we,
                                                     logit, emaxkey, E);

    // 6. exp + segment sum
    k_edge_exp<<<(E + B - 1) / B, B, 0, stream>>>(dst, emaxkey, logit, asum, E);

    // 7. reciprocal of sums
    k_asum_recip<<<((size_t)N * NH + B - 1) / B, B, 0, stream>>>(asum, N * NH);

    // 8. weighted aggregation (64 threads per edge)
    size_t aggT = (size_t)E * HD;
    k_aggregate<<<(aggT + B - 1) / B, B, 0, stream>>>(src, dst, logit, asum, featp, out, E);
}